// AtomicGraphEncoder_46986942218277
// MI455X (gfx1250) — compile-verified
//
#include <hip/hip_runtime.h>

typedef __attribute__((ext_vector_type(2))) float v2f;
typedef __attribute__((ext_vector_type(8))) float v8f;

#define N_NODES  50000   // divisible by 16 -> no M-edge handling in GEMM
#define N_EDGES  800000
#define N_GRAPHS 256
#define BN_EPS   1e-5f

// ---------------------------------------------------------------------------
// Edge scatter-add: one wave32 per edge, lanes sweep channels (coalesced row
// gather of h[src], coalesced atomic adds into agg[dst]).
// ---------------------------------------------------------------------------
__global__ void scatter_add_kernel(const float* __restrict__ H, int ldh,
                                   const int* __restrict__ src,
                                   const int* __restrict__ dst,
                                   int E, int K,
                                   float* __restrict__ Agg, int lda) {
  int w    = (blockIdx.x * blockDim.x + threadIdx.x) >> 5;
  int lane = threadIdx.x & 31;
  if (w >= E) return;
  int s = src[w];
  int d = dst[w];
  const float* hs = H + (size_t)s * ldh;
  float*       ad = Agg + (size_t)d * lda;
  for (int c = lane; c < K; c += 32) atomicAdd(&ad[c], hs[c]);
}

// ---------------------------------------------------------------------------
// Dual GEMM via fp32 WMMA:  Out = A0 @ W0 + A1 @ W1 + bias
// One wave per 16x16 output tile, K stepped by 4 (V_WMMA_F32_16X16X4_F32).
// Requirements exploited:
//   * M is a multiple of 16 (50000 and 256 both are) -> no row guards.
//   * lda is even and base pointers are 8B-aligned -> A fragment is one
//     aligned v2f (global_load_b64), unconditional.
//   * Out-of-range N columns are handled by CLAMPING the W column index:
//     those lanes compute unused D columns, and stores are guarded.
//   * Only the (rare) K%4 tail is predicated, peeled out of the main loop.
// EXEC is never manipulated around the WMMA (all 32 lanes always live).
// ---------------------------------------------------------------------------
__global__ void gemm_dual_wmma(const float* __restrict__ A0, int lda0,
                               const float* __restrict__ A1, int lda1,
                               const float* __restrict__ W0,
                               const float* __restrict__ W1,
                               const float* __restrict__ bias,
                               int K, int N,
                               float* __restrict__ Out, int ldo) {
  int mt   = blockIdx.x;
  int nt   = blockIdx.y;
  int lane = threadIdx.x & 31;
  int half = lane >> 4;   // 0: lanes 0-15, 1: lanes 16-31
  int l    = lane & 15;
  int row  = mt * 16 + l;            // A row owned by this lane (always < M)
  int col  = nt * 16 + l;            // B/D column owned by this lane
  int colc = col < N ? col : N - 1;  // clamped W column (unused lanes -> dup)

  v8f acc = {};
  const int Kmain = K & ~3;

  for (int pass = 0; pass < 2; ++pass) {
    const float* A   = pass ? A1 : A0;
    const float* W   = pass ? W1 : W0;
    int          lda = pass ? lda1 : lda0;
    if (!A) break;  // classifier path: single operand
    const float* arow = A + (size_t)row * lda;

#pragma unroll 2
    for (int k0 = 0; k0 < Kmain; k0 += 4) {
      int ka = k0 + half * 2;                       // even -> 8B aligned
      v2f a  = *(const v2f*)(arow + ka);            // one global_load_b64
      v2f b;
      b.x = W[(size_t)ka * N + colc];
      b.y = W[(size_t)(ka + 1) * N + colc];
      // V_WMMA_F32_16X16X4_F32: (neg_a, A, neg_b, B, c_mod, C, reuse_a, reuse_b)
      acc = __builtin_amdgcn_wmma_f32_16x16x4_f32(
          false, a, false, b, (short)0, acc, false, false);
    }

    if (Kmain < K) {  // peeled K tail (only cin=22 hits this)
      int ka = Kmain + half * 2;
      v2f a = {0.f, 0.f};
      v2f b = {0.f, 0.f};
      if (ka < K)     { a.x = arow[ka];     b.x = W[(size_t)ka * N + colc]; }
      if (ka + 1 < K) { a.y = arow[ka + 1]; b.y = W[(size_t)(ka + 1) * N + colc]; }
      acc = __builtin_amdgcn_wmma_f32_16x16x4_f32(
          false, a, false, b, (short)0, acc, false, false);
    }
  }

  if (col < N) {
    float bv = bias ? bias[col] : 0.f;
#pragma unroll
    for (int r = 0; r < 8; ++r) {
      int rr = mt * 16 + half * 8 + r;  // D layout: VGPR r -> M = 8*half + r
      Out[(size_t)rr * ldo + col] = acc[r] + bv;
    }
  }
}

// ---------------------------------------------------------------------------
// Per-channel batch statistics over the node dimension (one block / channel).
// The pre-BN buffer was just written, so column walks hit L2 (192 MB).
// ---------------------------------------------------------------------------
__global__ void stats_kernel(const float* __restrict__ Out, int ldo, int M,
                             float* __restrict__ mu, float* __restrict__ inv) {
  __shared__ float ssum[256];
  __shared__ float ssq[256];
  int c = blockIdx.x;
  float s = 0.f, q = 0.f;
  for (int r = threadIdx.x; r < M; r += blockDim.x) {
    float v = Out[(size_t)r * ldo + c];
    s += v;
    q += v * v;
  }
  ssum[threadIdx.x] = s;
  ssq[threadIdx.x]  = q;
  __syncthreads();
  for (int st = 128; st > 0; st >>= 1) {
    if ((int)threadIdx.x < st) {
      ssum[threadIdx.x] += ssum[threadIdx.x + st];
      ssq[threadIdx.x]  += ssq[threadIdx.x + st];
    }
    __syncthreads();
  }
  if (threadIdx.x == 0) {
    float m   = ssum[0] / (float)M;
    float var = ssq[0] / (float)M - m * m;  // biased variance, like jnp.var
    mu[c]  = m;
    inv[c] = rsqrtf(var + BN_EPS);
  }
}

// ---------------------------------------------------------------------------
// BN + ReLU, scattered into a column slice of the dense-block buffer.
// ---------------------------------------------------------------------------
__global__ void bn_relu_kernel(const float* __restrict__ Out, int ldo,
                               const float* __restrict__ mu,
                               const float* __restrict__ inv,
                               const float* __restrict__ gamma,
                               const float* __restrict__ beta,
                               float* __restrict__ Dst, int ldd, int coloff,
                               int M, int N) {
  long i = (long)blockIdx.x * blockDim.x + threadIdx.x;
  if (i >= (long)M * N) return;
  int r = (int)(i / N);
  int c = (int)(i % N);
  float v = (Out[(size_t)r * ldo + c] - mu[c]) * inv[c] * gamma[c] + beta[c];
  Dst[(size_t)r * ldd + coloff + c] = v > 0.f ? v : 0.f;
}

// ---------------------------------------------------------------------------
// Global mean pool: one wave per node, lanes sweep channels.
// ---------------------------------------------------------------------------
__global__ void pool_sum_kernel(const float* __restrict__ H, int ldh,
                                const int* __restrict__ batch, int M, int N,
                                float* __restrict__ pooled,
                                float* __restrict__ cnt) {
  int w    = (blockIdx.x * blockDim.x + threadIdx.x) >> 5;
  int lane = threadIdx.x & 31;
  if (w >= M) return;
  int b = batch[w];
  const float* hr = H + (size_t)w * ldh;
  float*       pr = pooled + (size_t)b * N;
  for (int c = lane; c < N; c += 32) atomicAdd(&pr[c], hr[c]);
  if (lane == 0) atomicAdd(&cnt[b], 1.f);
}

__global__ void pool_div_kernel(float* __restrict__ pooled,
                                const float* __restrict__ cnt, int G, int N) {
  int i = blockIdx.x * blockDim.x + threadIdx.x;
  if (i >= G * N) return;
  int g = i / N;
  float c = cnt[g];
  pooled[i] /= (c > 1.f ? c : 1.f);
}

// ---------------------------------------------------------------------------
// Host-side: one GraphConvBn layer = memset(agg) -> scatter -> WMMA GEMM ->
// stats -> BN+ReLU into the destination column slice.
// ---------------------------------------------------------------------------
static void run_conv(const float* inH, int ldh, int cin, int cout,
                     float* dstPtr, int ldd, int coloff,
                     const float* Wr, const float* br, const float* Wroot,
                     const float* gamma, const float* beta,
                     const int* src, const int* dst,
                     float* agg, float* outb, float* mu, float* inv,
                     hipStream_t stream) {
  hipMemsetAsync(agg, 0, (size_t)N_NODES * cin * sizeof(float), stream);
  int aggBlocks = (N_EDGES * 32 + 255) / 256;
  scatter_add_kernel<<<aggBlocks, 256, 0, stream>>>(inH, ldh, src, dst,
                                                    N_EDGES, cin, agg, cin);
  dim3 g(N_NODES / 16, (cout + 15) / 16);
  gemm_dual_wmma<<<g, 32, 0, stream>>>(agg, cin, inH, ldh, Wr, Wroot, br,
                                       cin, cout, outb, cout);
  stats_kernel<<<cout, 256, 0, stream>>>(outb, cout, N_NODES, mu, inv);
  long tot = (long)N_NODES * cout;
  bn_relu_kernel<<<(int)((tot + 255) / 256), 256, 0, stream>>>(
      outb, cout, mu, inv, gamma, beta, dstPtr, ldd, coloff, N_NODES, cout);
}

extern "C" void kernel_launch(void* const* d_in, const int* in_sizes, int n_in,
                              void* d_out, int out_size, void* d_ws,
                              size_t ws_size, hipStream_t stream) {
  (void)in_sizes; (void)n_in; (void)out_size; (void)ws_size;

  const float* x     = (const float*)d_in[0];     // (50000, 22)
  const int*   ei    = (const int*)d_in[1];       // (2, 800000)
  const int*   src   = ei;
  const int*   dstp  = ei + N_EDGES;
  const int*   batch = (const int*)d_in[2];       // (50000,)

  // Workspace carve-up (~271 MB of float32).
  float* ws = (float*)d_ws;
  size_t off = 0;
  float* hblk1 = ws + off; off += (size_t)N_NODES * 288;  // dense block 1
  float* hblk2 = ws + off; off += (size_t)N_NODES * 400;  // dense block 2
  float* tbuf  = ws + off; off += (size_t)N_NODES * 64;   // conv1 bottleneck
  float* agg   = ws + off; off += (size_t)N_NODES * 400;  // scatter-add dest
  float* outb  = ws + off; off += (size_t)N_NODES * 200;  // pre-BN output
  float* mu    = ws + off; off += 256;
  float* inv   = ws + off; off += 256;
  float* pooled= ws + off; off += (size_t)N_GRAPHS * 200;
  float* cnt   = ws + off; off += N_GRAPHS;
  float* hfin  = hblk1;  // block-1 buffer is dead after transition 1 -> reuse

  // params flattening: convs[l] = {Wr, br, Wroot, gamma, beta}, then cls_W/b
  auto P = [&](int l, int field) -> const float* {
    return (const float*)d_in[3 + 5 * l + field];
  };

  int idx = 0;

  // Stem: x(22) -> hblk1[:, 0:32)
  run_conv(x, 22, 22, 32, hblk1, 288, 0, P(idx, 0), P(idx, 1), P(idx, 2),
           P(idx, 3), P(idx, 4), src, dstp, agg, outb, mu, inv, stream);
  idx++;

  // Dense block 1 (c = 32, ld = 288)
  for (int i = 0; i < 8; ++i) {
    int cin1 = 32 + i * 32;
    run_conv(hblk1, 288, cin1, 64, tbuf, 64, 0, P(idx, 0), P(idx, 1),
             P(idx, 2), P(idx, 3), P(idx, 4), src, dstp, agg, outb, mu, inv,
             stream);
    idx++;
    run_conv(tbuf, 64, 64, 32, hblk1, 288, 32 + i * 32, P(idx, 0), P(idx, 1),
             P(idx, 2), P(idx, 3), P(idx, 4), src, dstp, agg, outb, mu, inv,
             stream);
    idx++;
  }
  // Transition 1: 288 -> 144 into hblk2[:, 0:144)
  run_conv(hblk1, 288, 288, 144, hblk2, 400, 0, P(idx, 0), P(idx, 1),
           P(idx, 2), P(idx, 3), P(idx, 4), src, dstp, agg, outb, mu, inv,
           stream);
  idx++;

  // Dense block 2 (c = 144, ld = 400)
  for (int i = 0; i < 8; ++i) {
    int cin1 = 144 + i * 32;
    run_conv(hblk2, 400, cin1, 64, tbuf, 64, 0, P(idx, 0), P(idx, 1),
             P(idx, 2), P(idx, 3), P(idx, 4), src, dstp, agg, outb, mu, inv,
             stream);
    idx++;
    run_conv(tbuf, 64, 64, 32, hblk2, 400, 144 + i * 32, P(idx, 0), P(idx, 1),
             P(idx, 2), P(idx, 3), P(idx, 4), src, dstp, agg, outb, mu, inv,
             stream);
    idx++;
  }
  // Transition 2: 400 -> 200 into hfin
  run_conv(hblk2, 400, 400, 200, hfin, 200, 0, P(idx, 0), P(idx, 1),
           P(idx, 2), P(idx, 3), P(idx, 4), src, dstp, agg, outb, mu, inv,
           stream);
  idx++;

  // Global mean pool over graphs
  hipMemsetAsync(pooled, 0, (size_t)N_GRAPHS * 200 * sizeof(float), stream);
  hipMemsetAsync(cnt, 0, N_GRAPHS * sizeof(float), stream);
  pool_sum_kernel<<<(N_NODES * 32 + 255) / 256, 256, 0, stream>>>(
      hfin, 200, batch, N_NODES, 200, pooled, cnt);
  pool_div_kernel<<<(N_GRAPHS * 200 + 255) / 256, 256, 0, stream>>>(
      pooled, cnt, N_GRAPHS, 200);

  // Classifier: (256,200) @ (200,96) + b -> d_out, via the same WMMA GEMM
  const float* clsW = (const float*)d_in[3 + 35 * 5];
  const float* clsb = (const float*)d_in[3 + 35 * 5 + 1];
  dim3 gc(N_GRAPHS / 16, (96 + 15) / 16);
  gemm_dual_wmma<<<gc, 32, 0, stream>>>(pooled, 200, nullptr, 0, clsW, nullptr,
                                        clsb, 200, 96, (float*)d_out, 96);
}